// FKA_Conv_Encoder_71975061946384
// MI455X (gfx1250) — compile-verified
//
#include <hip/hip_runtime.h>
#include <hip/hip_bf16.h>

typedef __attribute__((ext_vector_type(2))) float v2f;
typedef __attribute__((ext_vector_type(8))) float v8f;

constexpr int B    = 8;
constexpr int CIN  = 64;
constexpr int COUT = 64;
constexpr int NP   = 8192;   // points
constexpr int KN   = 16;     // neighbors / hidden channels

// ---------------------------------------------------------------------------
// Kernel 0: zero the stats accumulators
// ---------------------------------------------------------------------------
__global__ void k_zero(float* __restrict__ p, int n) {
    int i = blockIdx.x * 256 + threadIdx.x;
    if (i < n) p[i] = 0.f;
}

// ---------------------------------------------------------------------------
// Kernel 1: pts, dw (sigmoid weights), fc1 -> y1 [b][c16][N][K]; stats1 sums
// one thread per (b, n)
// ---------------------------------------------------------------------------
__global__ __launch_bounds__(256) void k_stage1(
    const float* __restrict__ pos, const float* __restrict__ sup,
    const int* __restrict__ nb, const float* __restrict__ w1,
    const float* __restrict__ alpha, const float* __restrict__ beta,
    float* __restrict__ dwv, float* __restrict__ y1, float* __restrict__ stats1)
{
    const int n = blockIdx.x * 256 + threadIdx.x;
    const int b = blockIdx.y;
    __shared__ float ssum[KN], ssq[KN];
    if (threadIdx.x < KN) { ssum[threadIdx.x] = 0.f; ssq[threadIdx.x] = 0.f; }
    __syncthreads();

    const float al = alpha[0], be = beta[0];
    const float sx = sup[(b*3+0)*NP + n];
    const float sy = sup[(b*3+1)*NP + n];
    const float sz = sup[(b*3+2)*NP + n];
    const int* nbp = nb + (b*NP + n)*KN;

    float p0[KN], p1[KN], p2[KN], dwr[KN];
    float s = 0.f;
#pragma unroll
    for (int k = 0; k < KN; ++k) {
        int i = nbp[k];
        float x0 = pos[(b*3+0)*NP + i] - sx;
        float x1 = pos[(b*3+1)*NP + i] - sy;
        float x2 = pos[(b*3+2)*NP + i] - sz;
        p0[k] = x0; p1[k] = x1; p2[k] = x2;
        float dd = sqrtf(x0*x0 + x1*x1 + x2*x2);
        float w  = 1.f / (1.f + __expf(al*dd - be));   // sigmoid(-a*d+b)
        dwr[k] = w; s += w;
    }
    float dws = s + ((s == 0.f) ? 1.f : 0.f) + 1e-6f;
    float sc  = (float)KN / dws;
    float* dwp = dwv + (b*NP + n)*KN;
#pragma unroll
    for (int k = 0; k < KN; ++k) { dwr[k] *= sc; dwp[k] = dwr[k]; }

#pragma unroll
    for (int c = 0; c < KN; ++c) {
        float wa = w1[c*3+0], wb = w1[c*3+1], wc = w1[c*3+2];
        float as = 0.f, aq = 0.f;
        float* yp = y1 + ((b*KN + c)*NP + n)*KN;
#pragma unroll
        for (int k = 0; k < KN; ++k) {
            float y = wa*p0[k] + wb*p1[k] + wc*p2[k];
            yp[k] = y; as += y; aq += y*y;
        }
        atomicAdd(&ssum[c], as);
        atomicAdd(&ssq[c],  aq);
    }
    __syncthreads();
    if (threadIdx.x < KN) {
        atomicAdd(&stats1[(b*KN + threadIdx.x)*2 + 0], ssum[threadIdx.x]);
        atomicAdd(&stats1[(b*KN + threadIdx.x)*2 + 1], ssq[threadIdx.x]);
    }
}

// ---------------------------------------------------------------------------
// Kernel 2/3: inorm(in)+relu, k-max with dw, concat, fc(32->16)
// FINAL=false: -> y2 [b][c16][N][K] + stats2
// FINAL=true : -> relu(fc3)*dw stored matF [b][N][K][j16]
// one thread per (b, n)
// ---------------------------------------------------------------------------
template <bool FINAL>
__global__ __launch_bounds__(256) void k_mid(
    const float* __restrict__ yin, const float* __restrict__ stats,
    const float* __restrict__ bnw, const float* __restrict__ bnb,
    const float* __restrict__ fw,   // [16][32]
    const float* __restrict__ dwv,
    float* __restrict__ yout, float* __restrict__ statsOut)
{
    const int n = blockIdx.x * 256 + threadIdx.x;
    const int b = blockIdx.y;
    __shared__ float aa[KN], bb[KN], wsh[KN*2*KN], osum[KN], osq[KN];
    if (threadIdx.x < KN) {
        int c = threadIdx.x;
        const float inv_cnt = 1.f / (float)(NP * KN);
        float mu  = stats[(b*KN + c)*2 + 0] * inv_cnt;
        float var = stats[(b*KN + c)*2 + 1] * inv_cnt - mu*mu;
        float inv = rsqrtf(var + 1e-5f);
        float a   = bnw[c] * inv;
        aa[c] = a; bb[c] = bnb[c] - mu*a;
        osum[c] = 0.f; osq[c] = 0.f;
    }
    for (int i = threadIdx.x; i < KN*2*KN; i += 256) wsh[i] = fw[i];
    __syncthreads();

    const float* dwp = dwv + (b*NP + n)*KN;

    // pass 1: per-channel weighted max over k (relu >= 0, dw > 0 -> init 0 ok)
    float mp[KN];
#pragma unroll
    for (int c = 0; c < KN; ++c) {
        const float* yp = yin + ((b*KN + c)*NP + n)*KN;
        float a = aa[c], bo = bb[c], m = 0.f;
#pragma unroll
        for (int k = 0; k < KN; ++k) {
            float v = fmaxf(a*yp[k] + bo, 0.f);
            m = fmaxf(m, v * dwp[k]);
        }
        mp[c] = m;
    }
    // k-independent half of the fc (second 16 inputs = broadcast max)
    float h[KN];
#pragma unroll
    for (int o = 0; o < KN; ++o) {
        float acc = 0.f;
#pragma unroll
        for (int c = 0; c < KN; ++c) acc += wsh[o*(2*KN) + KN + c] * mp[c];
        h[o] = acc;
    }

    float ls[KN], lq[KN];
    if (!FINAL) {
#pragma unroll
        for (int o = 0; o < KN; ++o) { ls[o] = 0.f; lq[o] = 0.f; }
    }
    for (int k = 0; k < KN; ++k) {   // rolled: keeps code size sane
        float acc[KN];
#pragma unroll
        for (int o = 0; o < KN; ++o) acc[o] = h[o];
#pragma unroll
        for (int c = 0; c < KN; ++c) {
            float v = fmaxf(aa[c]*yin[((b*KN + c)*NP + n)*KN + k] + bb[c], 0.f);
#pragma unroll
            for (int o = 0; o < KN; ++o) acc[o] += wsh[o*(2*KN) + c] * v;
        }
        if (FINAL) {
            float dk = dwp[k];
            float* op = yout + ((b*NP + n)*KN + k)*KN;   // [b][n][k][j]
#pragma unroll
            for (int o = 0; o < KN; ++o) op[o] = fmaxf(acc[o], 0.f) * dk;
        } else {
#pragma unroll
            for (int o = 0; o < KN; ++o) {
                float v = acc[o];
                yout[((b*KN + o)*NP + n)*KN + k] = v;
                ls[o] += v; lq[o] += v*v;
            }
        }
    }
    if (!FINAL) {
#pragma unroll
        for (int o = 0; o < KN; ++o) { atomicAdd(&osum[o], ls[o]); atomicAdd(&osq[o], lq[o]); }
        __syncthreads();
        if (threadIdx.x < KN) {
            atomicAdd(&statsOut[(b*KN + threadIdx.x)*2 + 0], osum[threadIdx.x]);
            atomicAdd(&statsOut[(b*KN + threadIdx.x)*2 + 1], osq[threadIdx.x]);
        }
    }
}

// ---------------------------------------------------------------------------
// Kernel 4 (WMMA, fp32): per wave, a tile of 16 points.
//   feats_ct (16c x 16j) = Xg_ct(16x16k) * M_n(16k x 16j)        (4x wmma K=4)
//   Out (64 x 16n)      += Wc_slice(64 x 256cj) * F(256cj x 16n)  (4x64 wmma)
// block = 128 threads (4 waves), covers 64 points; grid (NP/64, B)
// ---------------------------------------------------------------------------
__global__ __launch_bounds__(128) void k_wmma(
    const float* __restrict__ x, const int* __restrict__ nb,
    const float* __restrict__ matF, const float* __restrict__ cvw,
    float* __restrict__ out)
{
    __shared__ float cvs[COUT * 16 * 16];      // 64KB: Wc slice [o][c16][j16]
    __shared__ float Fbuf[4][16 * 16 * 16];    // 16KB/wave: [c][j][n]
    const int wave = threadIdx.x >> 5;
    const int lane = threadIdx.x & 31;
    const int l15  = lane & 15;
    const int hi   = lane >> 4;                // 0 or 1
    const int b    = blockIdx.y;
    const int n0   = blockIdx.x * 64 + wave * 16;
    float* F = Fbuf[wave];

    v8f acc[4] = {};                           // Out accumulators, 4 o-tiles

    for (int ct = 0; ct < 4; ++ct) {
        __syncthreads();                       // previous phase-2 done with cvs
        for (int i = threadIdx.x; i < COUT*256; i += 128) {
            int o = i >> 8, cj = i & 255;
            cvs[i] = cvw[o*1024 + ct*256 + cj];
        }

        // phase 1: per-point feats c-tile -> F LDS
        for (int nn = 0; nn < 16; ++nn) {
            const int n     = n0 + nn;
            const int nbase = (b*NP + n)*KN;
            v8f d = {};
#pragma unroll
            for (int kk = 0; kk < 16; kk += 4) {
                // A frag (16x4): row c = l15 (+ctile), VGPR0: K=kk+2*hi, VGPR1: +1
                int k0 = kk + hi*2;
                int c  = ct*16 + l15;
                int i0 = nb[nbase + k0];
                int i1 = nb[nbase + k0 + 1];
                v2f a;
                a.x = x[(b*CIN + c)*NP + i0];
                a.y = x[(b*CIN + c)*NP + i1];
                // B frag (4x16): col j = l15, rows k0 / k0+1 (matF row-major in j)
                v2f bf;
                bf.x = matF[(nbase + k0    )*16 + l15];
                bf.y = matF[(nbase + k0 + 1)*16 + l15];
                d = __builtin_amdgcn_wmma_f32_16x16x4_f32(
                        false, a, false, bf, (short)0, d, false, false);
            }
            // D (16c x 16j): VGPR r -> row r + 8*hi, col = l15; store F[c][j][n]
#pragma unroll
            for (int r = 0; r < 8; ++r)
                F[(r + hi*8)*256 + l15*16 + nn] = d[r];
        }
        __syncthreads();                       // cvs loaded (F is wave-private)

        // phase 2: Out += Wc_slice * F, contraction over 256 (c,j) pairs
#pragma unroll
        for (int t = 0; t < 4; ++t) {
            for (int kk = 0; kk < 256; kk += 4) {
                int r0 = kk + hi*2;
                int o  = t*16 + l15;
                v2f a, bf;
                a.x  = cvs[o*256 + r0];
                a.y  = cvs[o*256 + r0 + 1];
                bf.x = F[ r0     *16 + l15];
                bf.y = F[(r0 + 1)*16 + l15];
                acc[t] = __builtin_amdgcn_wmma_f32_16x16x4_f32(
                             false, a, false, bf, (short)0, acc[t], false, false);
            }
        }
    }
    // write out [b][o][n]
#pragma unroll
    for (int t = 0; t < 4; ++t)
#pragma unroll
        for (int r = 0; r < 8; ++r)
            out[(b*COUT + t*16 + r + hi*8)*NP + n0 + l15] = acc[t][r];
}

// ---------------------------------------------------------------------------
extern "C" void kernel_launch(void* const* d_in, const int* in_sizes, int n_in,
                              void* d_out, int out_size, void* d_ws, size_t ws_size,
                              hipStream_t stream) {
    (void)in_sizes; (void)n_in; (void)out_size; (void)ws_size;
    const float* x    = (const float*)d_in[0];
    const float* pos  = (const float*)d_in[1];
    const float* sup  = (const float*)d_in[2];
    const int*   nb   = (const int*)  d_in[3];
    const float* w1   = (const float*)d_in[4];
    const float* w2   = (const float*)d_in[5];
    const float* w3   = (const float*)d_in[6];
    const float* bn1w = (const float*)d_in[7];
    const float* bn1b = (const float*)d_in[8];
    const float* bn2w = (const float*)d_in[9];
    const float* bn2b = (const float*)d_in[10];
    const float* cvw  = (const float*)d_in[11];
    const float* al   = (const float*)d_in[12];
    const float* be   = (const float*)d_in[13];
    float* out = (float*)d_out;

    char* ws = (char*)d_ws;
    float* dwv    = (float*)(ws);                       //  4 MB
    float* stats1 = (float*)(ws + (size_t(4)  << 20));  //  1 KB
    float* stats2 = stats1 + 2*B*KN;
    float* y1     = (float*)(ws + (size_t(8)  << 20));  // 64 MB
    float* y2     = (float*)(ws + (size_t(72) << 20));  // 64 MB
    float* matF   = y1;                                 // y1 dead after stage 2

    k_zero<<<2, 256, 0, stream>>>(stats1, 2 * 2 * B * KN);
    dim3 g1(NP/256, B);
    k_stage1<<<g1, 256, 0, stream>>>(pos, sup, nb, w1, al, be, dwv, y1, stats1);
    k_mid<false><<<g1, 256, 0, stream>>>(y1, stats1, bn1w, bn1b, w2, dwv, y2, stats2);
    k_mid<true ><<<g1, 256, 0, stream>>>(y2, stats2, bn2w, bn2b, w3, dwv, matF, nullptr);
    dim3 g4(NP/64, B);
    k_wmma<<<g4, 128, 0, stream>>>(x, nb, matF, cvw, out);
}